// SimpleGNN_695784702108
// MI455X (gfx1250) — compile-verified
//
#include <hip/hip_runtime.h>
#include <hip/hip_bf16.h>

namespace {

constexpr int HDIM   = 64;   // hidden width
constexpr int FIN    = 128;  // input features
constexpr int NLAY   = 6;    // GCN layers
constexpr int NGRAPH = 64;   // graphs in batch
constexpr int NCLS   = 10;   // classes

typedef __attribute__((ext_vector_type(2))) float v2f;
typedef __attribute__((ext_vector_type(8))) float v8f;

// ---------------------------------------------------------------------------
// degree / normalization
// ---------------------------------------------------------------------------
__global__ void init_deg_pool(float* __restrict__ deg, int n,
                              float* __restrict__ pooled, int np) {
  int i = blockIdx.x * blockDim.x + threadIdx.x;
  if (i < n)  deg[i] = 1.0f;        // self-loop contributes 1 to every node
  if (i < np) pooled[i] = 0.0f;     // zero the pooling accumulator each call
}

__global__ void deg_accum(const int* __restrict__ dst, int e,
                          float* __restrict__ deg) {
  int i = blockIdx.x * blockDim.x + threadIdx.x;
  if (i < e) atomicAdd(&deg[dst[i]], 1.0f);
}

__global__ void make_disq(float* __restrict__ deg, int n) {
  int i = blockIdx.x * blockDim.x + threadIdx.x;
  if (i < n) {
    float d = deg[i];
    deg[i] = d > 0.0f ? rsqrtf(d) : 0.0f;   // in-place deg -> deg^{-1/2}
  }
}

// ---------------------------------------------------------------------------
// Dense GEMM on the CDNA5 matrix path: D = A[nrows x K] * W[K x 64]
//   V_WMMA_F32_16X16X4_F32.  One wave owns a 32-row tile x 64 columns:
//   two 16x16 row tiles share the four B fragments -> 8 WMMAs per
//   (2 global b64 + 4 ds b64) loads.
// W is staged in LDS pair-interleaved:  sW[(k/2)*128 + 2*c + (k&1)]
//   so each B fragment {W[ka][c], W[ka+1][c]} is one aligned ds_load_b64.
// A-frag (16x4 f32): lanes 0-15 -> K=k,k+1 ; lanes 16-31 -> K=k+2,k+3 (ka even
//   always, so the pair is one global b64).
// C/D: row = j + 8*(lane>=16), col = lane%16.
// EPI==0 (conv layers): write hW, and agg = hW*disq^2 + bias (self-loop+bias)
// EPI==1 (JK layer)   : write hW = relu(acc + bias)
// ---------------------------------------------------------------------------
template <int EPI>
__global__ __launch_bounds__(128)
void gcn_gemm_wmma(const float* __restrict__ A, int lda, int K,
                   const float* __restrict__ W,
                   const float* __restrict__ bias,
                   const float* __restrict__ disq,
                   float* __restrict__ hW,
                   float* __restrict__ agg,
                   int nrows) {
  extern __shared__ float sW[];                  // K/2 pairs x 128 floats
  for (int i = threadIdx.x; i < K * HDIM; i += blockDim.x) {
    const int k = i >> 6;                        // source row (K index)
    const int c = i & 63;                        // source column
    sW[((k >> 1) << 7) + (c << 1) + (k & 1)] = W[i];
  }
  __syncthreads();

  const int wave = threadIdx.x >> 5;
  const int lane = threadIdx.x & 31;
  const int tile = blockIdx.x * (blockDim.x >> 5) + wave;
  const int row0 = tile * 32;
  if (row0 >= nrows) return;                     // uniform per wave: EXEC all-1s

  const int half = lane >> 4;                    // 0: K pair lo, 1: K pair hi
  const int l16  = lane & 15;
  const float* arowLo = A + (long)(row0 + l16) * lda + 2 * half;
  const float* arowHi = arowLo + (long)16 * lda;
  const float* sWl    = sW + 2 * l16;            // per-lane B base

  v8f acc0 = {}, acc1 = {}, acc2 = {}, acc3 = {};   // rows row0 .. row0+15
  v8f acc4 = {}, acc5 = {}, acc6 = {}, acc7 = {};   // rows row0+16 .. row0+31
  for (int k = 0; k < K; k += 4) {
    const float* wp = sWl + (((k >> 1) + half) << 7);
    const v2f aLo = *reinterpret_cast<const v2f*>(arowLo + k);
    const v2f aHi = *reinterpret_cast<const v2f*>(arowHi + k);
    const v2f b0 = *reinterpret_cast<const v2f*>(wp);
    const v2f b1 = *reinterpret_cast<const v2f*>(wp + 32);
    const v2f b2 = *reinterpret_cast<const v2f*>(wp + 64);
    const v2f b3 = *reinterpret_cast<const v2f*>(wp + 96);
    acc0 = __builtin_amdgcn_wmma_f32_16x16x4_f32(false, aLo, false, b0, (short)0, acc0, false, false);
    acc1 = __builtin_amdgcn_wmma_f32_16x16x4_f32(false, aLo, false, b1, (short)0, acc1, false, false);
    acc2 = __builtin_amdgcn_wmma_f32_16x16x4_f32(false, aLo, false, b2, (short)0, acc2, false, false);
    acc3 = __builtin_amdgcn_wmma_f32_16x16x4_f32(false, aLo, false, b3, (short)0, acc3, false, false);
    acc4 = __builtin_amdgcn_wmma_f32_16x16x4_f32(false, aHi, false, b0, (short)0, acc4, false, false);
    acc5 = __builtin_amdgcn_wmma_f32_16x16x4_f32(false, aHi, false, b1, (short)0, acc5, false, false);
    acc6 = __builtin_amdgcn_wmma_f32_16x16x4_f32(false, aHi, false, b2, (short)0, acc6, false, false);
    acc7 = __builtin_amdgcn_wmma_f32_16x16x4_f32(false, aHi, false, b3, (short)0, acc7, false, false);
  }

#pragma unroll
  for (int j = 0; j < 8; ++j) {
    const int r0 = row0 + j + 8 * half;          // row tile 0
    const int r1 = r0 + 16;                      // row tile 1
    const long base0 = (long)r0 * HDIM;
    const long base1 = (long)r1 * HDIM;
    if constexpr (EPI == 0) {
      hW[base0 + l16     ] = acc0[j];
      hW[base0 + l16 + 16] = acc1[j];
      hW[base0 + l16 + 32] = acc2[j];
      hW[base0 + l16 + 48] = acc3[j];
      hW[base1 + l16     ] = acc4[j];
      hW[base1 + l16 + 16] = acc5[j];
      hW[base1 + l16 + 32] = acc6[j];
      hW[base1 + l16 + 48] = acc7[j];
      const float d0 = disq[r0], sl0 = d0 * d0;  // self-loop norm disq[i]^2
      const float d1 = disq[r1], sl1 = d1 * d1;
      agg[base0 + l16     ] = acc0[j] * sl0 + bias[l16     ];
      agg[base0 + l16 + 16] = acc1[j] * sl0 + bias[l16 + 16];
      agg[base0 + l16 + 32] = acc2[j] * sl0 + bias[l16 + 32];
      agg[base0 + l16 + 48] = acc3[j] * sl0 + bias[l16 + 48];
      agg[base1 + l16     ] = acc4[j] * sl1 + bias[l16     ];
      agg[base1 + l16 + 16] = acc5[j] * sl1 + bias[l16 + 16];
      agg[base1 + l16 + 32] = acc6[j] * sl1 + bias[l16 + 32];
      agg[base1 + l16 + 48] = acc7[j] * sl1 + bias[l16 + 48];
    } else {
      hW[base0 + l16     ] = fmaxf(acc0[j] + bias[l16     ], 0.0f);
      hW[base0 + l16 + 16] = fmaxf(acc1[j] + bias[l16 + 16], 0.0f);
      hW[base0 + l16 + 32] = fmaxf(acc2[j] + bias[l16 + 32], 0.0f);
      hW[base0 + l16 + 48] = fmaxf(acc3[j] + bias[l16 + 48], 0.0f);
      hW[base1 + l16     ] = fmaxf(acc4[j] + bias[l16     ], 0.0f);
      hW[base1 + l16 + 16] = fmaxf(acc5[j] + bias[l16 + 16], 0.0f);
      hW[base1 + l16 + 32] = fmaxf(acc6[j] + bias[l16 + 32], 0.0f);
      hW[base1 + l16 + 48] = fmaxf(acc7[j] + bias[l16 + 48], 0.0f);
    }
  }
}

// ---------------------------------------------------------------------------
// Edge propagation: agg[dst] += hW[src] * disq[src]*disq[dst]
// 16 lanes per edge, float4 per lane -> 256B coalesced gather, L2 f32 atomics.
// All per-layer arrays (25.6 MB) live in the 192 MB L2.
// ---------------------------------------------------------------------------
__global__ void gcn_scatter(const int* __restrict__ src, const int* __restrict__ dst,
                            int e, const float* __restrict__ disq,
                            const float* __restrict__ hW, float* __restrict__ agg) {
  const long total  = (long)e * 16;
  const long stride = (long)gridDim.x * blockDim.x;
  for (long gid = (long)blockIdx.x * blockDim.x + threadIdx.x; gid < total; gid += stride) {
    const int ei = (int)(gid >> 4);
    const int c  = ((int)gid & 15) << 2;
    const int s = src[ei], d = dst[ei];
    const float nrm = disq[s] * disq[d];
    const float4 v = *reinterpret_cast<const float4*>(hW + (long)s * HDIM + c);
    float* ap = agg + (long)d * HDIM + c;
    atomicAdd(ap + 0, v.x * nrm);
    atomicAdd(ap + 1, v.y * nrm);
    atomicAdd(ap + 2, v.z * nrm);
    atomicAdd(ap + 3, v.w * nrm);
  }
}

// ReLU(agg) into the layer's column slot of the JK concat buffer [N, 6*64]
__global__ void relu_to_xs(const float* __restrict__ agg, float* __restrict__ xs_slot,
                           int n) {
  const long total  = (long)n * 16;
  const long stride = (long)gridDim.x * blockDim.x;
  for (long gid = (long)blockIdx.x * blockDim.x + threadIdx.x; gid < total; gid += stride) {
    const int r = (int)(gid >> 4);
    const int c = ((int)gid & 15) << 2;
    const float4 v = *reinterpret_cast<const float4*>(agg + (long)r * HDIM + c);
    float4 o = { fmaxf(v.x, 0.0f), fmaxf(v.y, 0.0f), fmaxf(v.z, 0.0f), fmaxf(v.w, 0.0f) };
    *reinterpret_cast<float4*>(xs_slot + (long)r * (NLAY * HDIM) + c) = o;
  }
}

// global_add_pool over sorted batch ids -> pooled[64, 64] (16 KB, lives in L2)
__global__ void pool_batch(const int* __restrict__ batch, const float* __restrict__ hjk,
                           float* __restrict__ pooled, int n) {
  const long total  = (long)n * 16;
  const long stride = (long)gridDim.x * blockDim.x;
  for (long gid = (long)blockIdx.x * blockDim.x + threadIdx.x; gid < total; gid += stride) {
    const int r = (int)(gid >> 4);
    const int c = ((int)gid & 15) << 2;
    const int g = batch[r];
    const float4 v = *reinterpret_cast<const float4*>(hjk + (long)r * HDIM + c);
    float* pp = pooled + (long)g * HDIM + c;
    atomicAdd(pp + 0, v.x);
    atomicAdd(pp + 1, v.y);
    atomicAdd(pp + 2, v.z);
    atomicAdd(pp + 3, v.w);
  }
}

// Tiny head: relu(pooled @ W1 + b1) @ W2 + b2 -> softmax.  64x64 work, 1 block.
__global__ void head_kernel(const float* __restrict__ pooled,
                            const float* __restrict__ W1, const float* __restrict__ b1,
                            const float* __restrict__ W2, const float* __restrict__ b2,
                            float* __restrict__ out) {
  __shared__ float ht[NGRAPH * HDIM];
  __shared__ float lg[NGRAPH * NCLS];
  for (int i = threadIdx.x; i < NGRAPH * HDIM; i += blockDim.x) {
    const int g = i >> 6, j = i & 63;
    float s = b1[j];
    for (int k = 0; k < HDIM; ++k) s += pooled[g * HDIM + k] * W1[k * HDIM + j];
    ht[i] = fmaxf(s, 0.0f);
  }
  __syncthreads();
  for (int i = threadIdx.x; i < NGRAPH * NCLS; i += blockDim.x) {
    const int g = i / NCLS, c = i % NCLS;
    float s = b2[c];
    for (int j = 0; j < HDIM; ++j) s += ht[g * HDIM + j] * W2[j * NCLS + c];
    lg[i] = s;
  }
  __syncthreads();
  if (threadIdx.x < NGRAPH) {
    const int g = threadIdx.x;
    float m = -INFINITY;
    for (int c = 0; c < NCLS; ++c) m = fmaxf(m, lg[g * NCLS + c]);
    float e[NCLS], sum = 0.0f;
    for (int c = 0; c < NCLS; ++c) { e[c] = __expf(lg[g * NCLS + c] - m); sum += e[c]; }
    const float inv = 1.0f / sum;
    for (int c = 0; c < NCLS; ++c) out[g * NCLS + c] = e[c] * inv;
  }
}

}  // namespace

// ---------------------------------------------------------------------------
// inputs: 0 x[N,128] 1 edge_index[2,E] 2 batch[N] 3 W0 4 b0 5 Wh[5,64,64]
//         6 bh[5,64] 7 jkW[384,64] 8 jkb 9 W1 10 b1 11 W2 12 b2
// workspace (floats): disq[N] | hW[N*64] | agg[N*64] | xs[N*384] | pooled[4096]
//   ~206 MB total
// ---------------------------------------------------------------------------
extern "C" void kernel_launch(void* const* d_in, const int* in_sizes, int n_in,
                              void* d_out, int out_size, void* d_ws, size_t ws_size,
                              hipStream_t stream) {
  const int N = in_sizes[0] / FIN;
  const int E = in_sizes[1] / 2;

  const float* x   = (const float*)d_in[0];
  const int*   ei  = (const int*)d_in[1];
  const int*   bat = (const int*)d_in[2];
  const float* W0  = (const float*)d_in[3];
  const float* b0  = (const float*)d_in[4];
  const float* Wh  = (const float*)d_in[5];
  const float* bh  = (const float*)d_in[6];
  const float* jkW = (const float*)d_in[7];
  const float* jkb = (const float*)d_in[8];
  const float* W1  = (const float*)d_in[9];
  const float* b1  = (const float*)d_in[10];
  const float* W2  = (const float*)d_in[11];
  const float* b2  = (const float*)d_in[12];
  const int* src = ei;
  const int* dst = ei + E;

  float* ws = (float*)d_ws;
  size_t off = 0;
  float* disq = ws;                 off += ((size_t)N + 127) & ~(size_t)127;
  float* hW   = ws + off;           off += (size_t)N * HDIM;
  float* agg  = ws + off;           off += (size_t)N * HDIM;
  float* xs   = ws + off;           off += (size_t)N * HDIM * NLAY;
  float* pooled = ws + off;         off += NGRAPH * HDIM;

  const int T = 256;
  // normalization: deg = 1 + in-degree, then deg^{-1/2} in place
  init_deg_pool<<<(N + T - 1) / T, T, 0, stream>>>(disq, N, pooled, NGRAPH * HDIM);
  deg_accum<<<(E + T - 1) / T, T, 0, stream>>>(dst, E, disq);
  make_disq<<<(N + T - 1) / T, T, 0, stream>>>(disq, N);

  const int tiles      = (N + 31) / 32;          // 3125 (N is a multiple of 32)
  const int gemmBlocks = (tiles + 3) / 4;        // 4 waves (32-row tiles) / block
  const int ewBlocks   = 4096;                   // grid-stride elementwise/scatter

  // layer 0: x @ W0  (K = 128)
  gcn_gemm_wmma<0><<<gemmBlocks, 128, FIN * HDIM * sizeof(float), stream>>>(
      x, FIN, FIN, W0, b0, disq, hW, agg, N);
  gcn_scatter<<<2 * ewBlocks, T, 0, stream>>>(src, dst, E, disq, hW, agg);
  relu_to_xs<<<ewBlocks, T, 0, stream>>>(agg, xs + 0 * HDIM, N);

  // layers 1..5: h @ Wh[l-1]  (K = 64, input read from JK slot l-1, stride 384)
  for (int l = 1; l < NLAY; ++l) {
    gcn_gemm_wmma<0><<<gemmBlocks, 128, HDIM * HDIM * sizeof(float), stream>>>(
        xs + (l - 1) * HDIM, NLAY * HDIM, HDIM,
        Wh + (size_t)(l - 1) * HDIM * HDIM, bh + (l - 1) * HDIM,
        disq, hW, agg, N);
    gcn_scatter<<<2 * ewBlocks, T, 0, stream>>>(src, dst, E, disq, hW, agg);
    relu_to_xs<<<ewBlocks, T, 0, stream>>>(agg, xs + l * HDIM, N);
  }

  // JK: relu(hcat @ jkW + jkb) -> reuse hW buffer  (K = 384)
  gcn_gemm_wmma<1><<<gemmBlocks, 128, NLAY * HDIM * HDIM * sizeof(float), stream>>>(
      xs, NLAY * HDIM, NLAY * HDIM, jkW, jkb, nullptr, hW, nullptr, N);

  pool_batch<<<ewBlocks, T, 0, stream>>>(bat, hW, pooled, N);
  head_kernel<<<1, T, 0, stream>>>(pooled, W1, b1, W2, b2, (float*)d_out);
}